// DecoderBlock_28166395527956
// MI455X (gfx1250) — compile-verified
//
#include <hip/hip_runtime.h>

typedef __attribute__((ext_vector_type(2))) float v2f;
typedef __attribute__((ext_vector_type(8))) float v8f;

#define WMMA_F32(a, b, c) \
    __builtin_amdgcn_wmma_f32_16x16x4_f32(false, (a), false, (b), (short)0, (c), false, false)

// Problem constants (hard-coded in the reference)
#define BATCH 8
#define SEQ   2048
#define DMODEL 64
#define NHEAD 4
#define HID   256
#define ROWS  (BATCH * SEQ)     // 16384
#define MTILES (ROWS / 16)      // 1024

// ---------------------------------------------------------------------------
// LayerNorm: one wave per row (D=64 -> 2 elements per lane), shuffle reduce.
// ---------------------------------------------------------------------------
__global__ void ln_kernel(const float* __restrict__ x,
                          const float* __restrict__ g,
                          const float* __restrict__ b,
                          float* __restrict__ out, int nrows) {
    int wid  = (blockIdx.x * blockDim.x + threadIdx.x) >> 5;
    int lane = threadIdx.x & 31;
    if (wid >= nrows) return;
    const float* row = x + (size_t)wid * DMODEL;
    float a0 = row[lane], a1 = row[lane + 32];
    float s = a0 + a1, ss = a0 * a0 + a1 * a1;
    for (int off = 16; off >= 1; off >>= 1) {
        s  += __shfl_xor(s, off);
        ss += __shfl_xor(ss, off);
    }
    float mu  = s * (1.0f / DMODEL);
    float var = ss * (1.0f / DMODEL) - mu * mu;
    float r   = rsqrtf(var + 1e-5f);
    float* orow = out + (size_t)wid * DMODEL;
    orow[lane]      = (a0 - mu) * r * g[lane]      + b[lane];
    orow[lane + 32] = (a1 - mu) * r * g[lane + 32] + b[lane + 32];
}

// ---------------------------------------------------------------------------
// QKV projection: y = xn @ W[h].T for 12 (head, {q,k,v}) combos.
// One wave computes a 16x64 output tile using V_WMMA_F32_16X16X4_F32.
// A layout (16x4 f32): lanes 0-15 -> K=k0,k0+1 ; lanes 16-31 -> K=k0+2,k0+3.
// B layout mirrored: lane gives N, half gives K pair. B[k][n] = W[n][k].
// ---------------------------------------------------------------------------
__global__ void qkv_kernel(const float* __restrict__ xn,
                           const float* __restrict__ Wq,
                           const float* __restrict__ Wk,
                           const float* __restrict__ Wv,
                           float* __restrict__ qb,
                           float* __restrict__ kb,
                           float* __restrict__ vb) {
    int wave = threadIdx.x >> 5, lane = threadIdx.x & 31;
    int tile = blockIdx.x * 4 + wave;           // 0..1023
    int mat  = blockIdx.y;                      // 0..11
    int h = mat / 3, kind = mat % 3;
    const float* W = (kind == 0 ? Wq : kind == 1 ? Wk : Wv) + (size_t)h * DMODEL * DMODEL;
    float* out = (kind == 0 ? qb : kind == 1 ? kb : vb) + (size_t)h * ROWS * DMODEL;

    int mloc = lane & 15;
    int koff = (lane < 16) ? 0 : 2;
    const float* arow = xn + (size_t)(tile * 16 + mloc) * DMODEL;
    v2f A[16];
    for (int kk = 0; kk < 16; kk++)
        A[kk] = *(const v2f*)(arow + kk * 4 + koff);

    for (int nt = 0; nt < 4; nt++) {
        v8f acc = {};
        for (int kk = 0; kk < 16; kk++) {
            const float* wrow = W + (size_t)(nt * 16 + mloc) * DMODEL + kk * 4 + koff;
            v2f Bf = *(const v2f*)wrow;
            acc = WMMA_F32(A[kk], Bf, acc);
        }
        int base_m = tile * 16 + ((lane < 16) ? 0 : 8);
        int col    = nt * 16 + mloc;
        for (int r = 0; r < 8; r++)
            out[(size_t)(base_m + r) * DMODEL + col] = acc[r];
    }
}

// ---------------------------------------------------------------------------
// Flash attention, causal. One wave per (b, h, 16-query-row tile).
// Scores via WMMA f32; online softmax with half-wave shuffle reductions
// (C layout: VGPR r / lanes 0-15 = row r, lanes 16-31 = row r+8).
// P (C layout) -> A layout transpose goes through per-wave LDS (in-order DS).
// ---------------------------------------------------------------------------
__global__ void attn_kernel(const float* __restrict__ qbuf,
                            const float* __restrict__ kbuf,
                            const float* __restrict__ vbuf,
                            float* __restrict__ attnout) {
    __shared__ float Pbuf[4][16][17];
    int wave = threadIdx.x >> 5, lane = threadIdx.x & 31;
    int i  = blockIdx.x * 4 + wave;             // query row-tile 0..127
    int bh = blockIdx.y;                        // 0..31
    int h  = bh & 3, b = bh >> 2;

    size_t base = ((size_t)h * ROWS + (size_t)b * SEQ) * DMODEL;
    const float* Q = qbuf + base;
    const float* K = kbuf + base;
    const float* V = vbuf + base;

    int mloc  = lane & 15;
    int koff  = (lane < 16) ? 0 : 2;
    int mbase = (lane < 16) ? 0 : 8;
    const float scale = 0.125f;                 // 1/sqrt(64)

    const float* qrow = Q + (size_t)(i * 16 + mloc) * DMODEL;
    v2f qA[16];
    for (int kk = 0; kk < 16; kk++) {
        v2f t = *(const v2f*)(qrow + kk * 4 + koff);
        qA[kk] = t * scale;
    }

    v8f O[4] = {{}, {}, {}, {}};
    float mrun[8], lrun[8];
    for (int r = 0; r < 8; r++) { mrun[r] = -1e30f; lrun[r] = 0.0f; }

    for (int j = 0; j <= i; j++) {
        // --- scores: S = (Q*scale) @ K_j^T -------------------------------
        v8f sc = {};
        const float* krow = K + (size_t)(j * 16 + mloc) * DMODEL;
        for (int kk = 0; kk < 16; kk++) {
            v2f bf = *(const v2f*)(krow + kk * 4 + koff);
            sc = WMMA_F32(qA[kk], bf, sc);
        }
        if (j == i) {                            // causal mask on diagonal tile
            for (int r = 0; r < 8; r++)
                if (mloc > r + mbase) sc[r] = -1e30f;
        }
        // --- online softmax per row --------------------------------------
        for (int r = 0; r < 8; r++) {
            float mx = sc[r];
            mx = fmaxf(mx, __shfl_xor(mx, 1));
            mx = fmaxf(mx, __shfl_xor(mx, 2));
            mx = fmaxf(mx, __shfl_xor(mx, 4));
            mx = fmaxf(mx, __shfl_xor(mx, 8));
            float mnew  = fmaxf(mrun[r], mx);
            float alpha = __expf(mrun[r] - mnew);
            float p     = __expf(sc[r] - mnew);
            float rs = p;
            rs += __shfl_xor(rs, 1);
            rs += __shfl_xor(rs, 2);
            rs += __shfl_xor(rs, 4);
            rs += __shfl_xor(rs, 8);
            lrun[r] = lrun[r] * alpha + rs;
            mrun[r] = mnew;
            O[0][r] *= alpha; O[1][r] *= alpha; O[2][r] *= alpha; O[3][r] *= alpha;
            Pbuf[wave][r + mbase][mloc] = p;     // stage P in C layout
        }
        // --- O += P @ V_j (read P back in A layout from LDS) -------------
        for (int kk = 0; kk < 4; kk++) {
            v2f aP;
            aP.x = Pbuf[wave][mloc][kk * 4 + koff];
            aP.y = Pbuf[wave][mloc][kk * 4 + koff + 1];
            const float* vrow0 = V + (size_t)(j * 16 + kk * 4 + koff) * DMODEL;
            const float* vrow1 = vrow0 + DMODEL;
            for (int nt = 0; nt < 4; nt++) {
                int n = nt * 16 + mloc;
                v2f bf; bf.x = vrow0[n]; bf.y = vrow1[n];
                O[nt] = WMMA_F32(aP, bf, O[nt]);
            }
        }
    }

    // epilogue: normalize + store concat-heads [B*S, 256]
    for (int r = 0; r < 8; r++) {
        float inv = 1.0f / lrun[r];
        int row = b * SEQ + i * 16 + r + mbase;
        for (int nt = 0; nt < 4; nt++)
            attnout[(size_t)row * HID + h * DMODEL + nt * 16 + mloc] = O[nt][r] * inv;
    }
}

// ---------------------------------------------------------------------------
// Output projection + residual: y = x + attnout @ Wo.T   (K=256, N=64)
// ---------------------------------------------------------------------------
__global__ void wo_kernel(const float* __restrict__ attnout,
                          const float* __restrict__ Wo,
                          const float* __restrict__ x,
                          float* __restrict__ y) {
    int wave = threadIdx.x >> 5, lane = threadIdx.x & 31;
    int tile = blockIdx.x * 4 + wave;
    int mloc = lane & 15;
    int koff = (lane < 16) ? 0 : 2;
    const float* arow = attnout + (size_t)(tile * 16 + mloc) * HID;

    v8f acc[4] = {{}, {}, {}, {}};
    for (int kk = 0; kk < 64; kk++) {
        v2f A = *(const v2f*)(arow + kk * 4 + koff);
        for (int nt = 0; nt < 4; nt++) {
            const float* wrow = Wo + (size_t)(nt * 16 + mloc) * HID + kk * 4 + koff;
            v2f Bf = *(const v2f*)wrow;
            acc[nt] = WMMA_F32(A, Bf, acc[nt]);
        }
    }
    int base_m = tile * 16 + ((lane < 16) ? 0 : 8);
    for (int nt = 0; nt < 4; nt++) {
        int col = nt * 16 + mloc;
        for (int r = 0; r < 8; r++) {
            size_t idx = (size_t)(base_m + r) * DMODEL + col;
            y[idx] = x[idx] + acc[nt][r];
        }
    }
}

// ---------------------------------------------------------------------------
// FC1: ffh = relu(h2 @ fc1_w.T + fc1_b)   (K=64, N=256; grid.y picks 64 cols)
// ---------------------------------------------------------------------------
__global__ void fc1_kernel(const float* __restrict__ h2,
                           const float* __restrict__ fc1_w,
                           const float* __restrict__ fc1_b,
                           float* __restrict__ ffh) {
    int wave = threadIdx.x >> 5, lane = threadIdx.x & 31;
    int tile = blockIdx.x * 4 + wave;
    int ng   = blockIdx.y;                      // 0..3 -> cols [ng*64, ng*64+64)
    int mloc = lane & 15;
    int koff = (lane < 16) ? 0 : 2;
    const float* arow = h2 + (size_t)(tile * 16 + mloc) * DMODEL;

    v2f A[16];
    for (int kk = 0; kk < 16; kk++)
        A[kk] = *(const v2f*)(arow + kk * 4 + koff);

    v8f acc[4] = {{}, {}, {}, {}};
    for (int kk = 0; kk < 16; kk++) {
        for (int nt = 0; nt < 4; nt++) {
            int n = ng * 64 + nt * 16 + mloc;
            const float* wrow = fc1_w + (size_t)n * DMODEL + kk * 4 + koff;
            v2f Bf = *(const v2f*)wrow;
            acc[nt] = WMMA_F32(A[kk], Bf, acc[nt]);
        }
    }
    int base_m = tile * 16 + ((lane < 16) ? 0 : 8);
    for (int nt = 0; nt < 4; nt++) {
        int col = ng * 64 + nt * 16 + mloc;
        float bias = fc1_b[col];
        for (int r = 0; r < 8; r++)
            ffh[(size_t)(base_m + r) * HID + col] = fmaxf(acc[nt][r] + bias, 0.0f);
    }
}

// ---------------------------------------------------------------------------
// FC2 + bias + residual: out = y + ffh @ fc2_w.T + fc2_b  (K=256, N=64)
// ---------------------------------------------------------------------------
__global__ void fc2_kernel(const float* __restrict__ ffh,
                           const float* __restrict__ fc2_w,
                           const float* __restrict__ fc2_b,
                           const float* __restrict__ y,
                           float* __restrict__ out) {
    int wave = threadIdx.x >> 5, lane = threadIdx.x & 31;
    int tile = blockIdx.x * 4 + wave;
    int mloc = lane & 15;
    int koff = (lane < 16) ? 0 : 2;
    const float* arow = ffh + (size_t)(tile * 16 + mloc) * HID;

    v8f acc[4] = {{}, {}, {}, {}};
    for (int kk = 0; kk < 64; kk++) {
        v2f A = *(const v2f*)(arow + kk * 4 + koff);
        for (int nt = 0; nt < 4; nt++) {
            const float* wrow = fc2_w + (size_t)(nt * 16 + mloc) * HID + kk * 4 + koff;
            v2f Bf = *(const v2f*)wrow;
            acc[nt] = WMMA_F32(A, Bf, acc[nt]);
        }
    }
    int base_m = tile * 16 + ((lane < 16) ? 0 : 8);
    for (int nt = 0; nt < 4; nt++) {
        int col = nt * 16 + mloc;
        float bias = fc2_b[col];
        for (int r = 0; r < 8; r++) {
            size_t idx = (size_t)(base_m + r) * DMODEL + col;
            out[idx] = y[idx] + acc[nt][r] + bias;
        }
    }
}

// ---------------------------------------------------------------------------
extern "C" void kernel_launch(void* const* d_in, const int* in_sizes, int n_in,
                              void* d_out, int out_size, void* d_ws, size_t ws_size,
                              hipStream_t stream) {
    const float* x     = (const float*)d_in[0];
    const float* Wq    = (const float*)d_in[1];
    const float* Wk    = (const float*)d_in[2];
    const float* Wv    = (const float*)d_in[3];
    const float* Wo    = (const float*)d_in[4];
    const float* fc1_w = (const float*)d_in[5];
    const float* fc1_b = (const float*)d_in[6];
    const float* fc2_w = (const float*)d_in[7];
    const float* fc2_b = (const float*)d_in[8];
    const float* ln1_g = (const float*)d_in[9];
    const float* ln1_b = (const float*)d_in[10];
    const float* ln2_g = (const float*)d_in[11];
    const float* ln2_b = (const float*)d_in[12];
    float* out = (float*)d_out;

    // workspace carve-up (fp32 elements)
    float* ws      = (float*)d_ws;
    float* xn      = ws;                                  //  1,048,576
    float* qb      = xn      + (size_t)ROWS * DMODEL;     //  4,194,304
    float* kb      = qb      + (size_t)NHEAD * ROWS * DMODEL;
    float* vb      = kb      + (size_t)NHEAD * ROWS * DMODEL;
    float* attnout = vb      + (size_t)NHEAD * ROWS * DMODEL;  // 4,194,304
    float* yb      = attnout + (size_t)ROWS * HID;        //  1,048,576
    float* h2      = yb      + (size_t)ROWS * DMODEL;     //  1,048,576
    float* ffh     = h2      + (size_t)ROWS * DMODEL;     //  4,194,304

    // 1. LayerNorm1
    ln_kernel<<<dim3(ROWS / 8), 256, 0, stream>>>(x, ln1_g, ln1_b, xn, ROWS);
    // 2. QKV projections (12 GEMMs)
    qkv_kernel<<<dim3(MTILES / 4, 12), 128, 0, stream>>>(xn, Wq, Wk, Wv, qb, kb, vb);
    // 3. causal flash attention
    attn_kernel<<<dim3((SEQ / 16) / 4, BATCH * NHEAD), 128, 0, stream>>>(qb, kb, vb, attnout);
    // 4. output projection + residual
    wo_kernel<<<dim3(MTILES / 4), 128, 0, stream>>>(attnout, Wo, x, yb);
    // 5. LayerNorm2
    ln_kernel<<<dim3(ROWS / 8), 256, 0, stream>>>(yb, ln2_g, ln2_b, h2, ROWS);
    // 6. FC1 + bias + ReLU
    fc1_kernel<<<dim3(MTILES / 4, 4), 128, 0, stream>>>(h2, fc1_w, fc1_b, ffh);
    // 7. FC2 + bias + residual -> final output
    fc2_kernel<<<dim3(MTILES / 4), 128, 0, stream>>>(ffh, fc2_w, fc2_b, yb, out);
}